// Confusion_Matrix_57251914056111
// MI455X (gfx1250) — compile-verified
//
#include <hip/hip_runtime.h>

// Confusion matrix: out[b, i, j] = #pixels with yp==i && y==j, per batch.
// B=8, N=2048*2048 per batch, C=21 -> 441 bins (padded to 448 = 28*16).

#define C_CLASSES 21
#define CC        441
#define CCP       448   // 28 * 16, WMMA-tile friendly

typedef __attribute__((ext_vector_type(2))) float v2f;
typedef __attribute__((ext_vector_type(8))) float v8f;

// ---------------------------------------------------------------------------
// Kernel 1: per-workgroup privatized LDS histogram, streamed b128 loads.
// Writes float partial histograms to workspace: partials[b][wg][CCP].
// ---------------------------------------------------------------------------
__global__ __launch_bounds__(256) void hist_partial_kernel(
    const int* __restrict__ yp, const int* __restrict__ y,
    float* __restrict__ partials, int Np, int P)
{
    __shared__ unsigned hist[CCP];
    const int tid = threadIdx.x;
    const int wg  = blockIdx.x;      // 0..P-1
    const int b   = blockIdx.y;      // batch

    for (int i = tid; i < CCP; i += 256) hist[i] = 0u;
    __syncthreads();

    const size_t base = (size_t)b * (size_t)Np;
    const int4* __restrict__ yp4 = (const int4*)(yp + base);
    const int4* __restrict__ y4  = (const int4*)(y  + base);
    const int Nv     = Np >> 2;          // int4 elements per batch
    const int stride = P * 256;

    for (int i = wg * 256 + tid; i < Nv; i += stride) {
        const int4 p = yp4[i];           // global_load_b128
        const int4 q = y4[i];
        atomicAdd(&hist[(unsigned)(p.x * C_CLASSES + q.x)], 1u);  // ds_add_u32
        atomicAdd(&hist[(unsigned)(p.y * C_CLASSES + q.y)], 1u);
        atomicAdd(&hist[(unsigned)(p.z * C_CLASSES + q.z)], 1u);
        atomicAdd(&hist[(unsigned)(p.w * C_CLASSES + q.w)], 1u);
    }
    // scalar tail (Np not divisible by 4)
    for (int i = (Nv << 2) + wg * 256 + tid; i < Np; i += stride) {
        atomicAdd(&hist[(unsigned)(yp[base + i] * C_CLASSES + y[base + i])], 1u);
    }

    __syncthreads();
    float* __restrict__ dst = partials + ((size_t)b * (size_t)P + (size_t)wg) * CCP;
    for (int i = tid; i < CCP; i += 256) dst[i] = (float)hist[i];
}

// ---------------------------------------------------------------------------
// Kernel 2: reduce P partials per batch with V_WMMA_F32_16X16X4_F32.
//   D(16x16) += A(16x4, all ones) x B(4x16 slice of partials)
// Since A is all ones, the K-permutation inside B's VGPR layout is irrelevant:
// D[m][n] = sum_k B[k][n]. One wave32 per batch; EXEC is all ones for WMMA.
// D layout: VGPR0, lanes 0-15 -> row M=0, col N=lane.
// ---------------------------------------------------------------------------
__global__ __launch_bounds__(32) void wmma_reduce_kernel(
    const float* __restrict__ partials, float* __restrict__ out, int P)
{
    const int b    = blockIdx.x;
    const int lane = threadIdx.x;        // wave32
    const int n    = lane & 15;
    const int half = lane >> 4;

    const v2f ones = {1.0f, 1.0f};
    const float* __restrict__ pb = partials + (size_t)b * (size_t)P * CCP;

    for (int t = 0; t < CCP / 16; ++t) {
        v8f acc = {};
        const int col = t * 16 + n;
        for (int kk = 0; kk < P; kk += 4) {
            // Assign the 4 K-values bijectively across (vgpr, half-wave);
            // exact mapping doesn't matter because A == 1 everywhere.
            const int k0 = kk + (half ? 2 : 0);
            const int k1 = k0 + 1;
            v2f bm;
            bm.x = pb[(size_t)k0 * CCP + col];
            bm.y = pb[(size_t)k1 * CCP + col];
            acc = __builtin_amdgcn_wmma_f32_16x16x4_f32(
                /*neg_a=*/false, ones, /*neg_b=*/false, bm,
                /*c_mod=*/(short)0, acc, /*reuse_a=*/false, /*reuse_b=*/false);
        }
        if (lane < 16 && col < CC) {
            out[(size_t)b * CC + col] = acc[0];   // row M=0 of D
        }
    }
}

// ---------------------------------------------------------------------------
// Fallback (workspace too small): zero output then LDS-hist + global f32 atomics.
// ---------------------------------------------------------------------------
__global__ void zero_out_kernel(float* __restrict__ out, int n)
{
    int i = blockIdx.x * 256 + threadIdx.x;
    if (i < n) out[i] = 0.0f;
}

__global__ __launch_bounds__(256) void hist_atomic_kernel(
    const int* __restrict__ yp, const int* __restrict__ y,
    float* __restrict__ out, int Np, int P)
{
    __shared__ unsigned hist[CCP];
    const int tid = threadIdx.x;
    const int wg  = blockIdx.x;
    const int b   = blockIdx.y;

    for (int i = tid; i < CCP; i += 256) hist[i] = 0u;
    __syncthreads();

    const size_t base = (size_t)b * (size_t)Np;
    const int4* __restrict__ yp4 = (const int4*)(yp + base);
    const int4* __restrict__ y4  = (const int4*)(y  + base);
    const int Nv     = Np >> 2;
    const int stride = P * 256;

    for (int i = wg * 256 + tid; i < Nv; i += stride) {
        const int4 p = yp4[i];
        const int4 q = y4[i];
        atomicAdd(&hist[(unsigned)(p.x * C_CLASSES + q.x)], 1u);
        atomicAdd(&hist[(unsigned)(p.y * C_CLASSES + q.y)], 1u);
        atomicAdd(&hist[(unsigned)(p.z * C_CLASSES + q.z)], 1u);
        atomicAdd(&hist[(unsigned)(p.w * C_CLASSES + q.w)], 1u);
    }
    for (int i = (Nv << 2) + wg * 256 + tid; i < Np; i += stride) {
        atomicAdd(&hist[(unsigned)(yp[base + i] * C_CLASSES + y[base + i])], 1u);
    }

    __syncthreads();
    for (int i = tid; i < CC; i += 256) {
        const unsigned c = hist[i];
        if (c) atomicAdd(&out[(size_t)b * CC + i], (float)c);  // global_atomic_add_f32
    }
}

// ---------------------------------------------------------------------------
extern "C" void kernel_launch(void* const* d_in, const int* in_sizes, int n_in,
                              void* d_out, int out_size, void* d_ws, size_t ws_size,
                              hipStream_t stream)
{
    const int* yp = (const int*)d_in[0];
    const int* y  = (const int*)d_in[1];
    // d_in[2] = res (all zeros, fully overwritten by reference -> ignored)
    // d_in[3] = n_classes scalar on device; reference fixes C=21.
    float* out = (float*)d_out;

    const int B  = out_size / CC;        // 8
    const int Np = (B > 0) ? (in_sizes[0] / B) : 0;   // 4,194,304

    // Choose #workgroups per batch from available workspace (deterministic).
    const size_t bytes_per_partial = sizeof(float) * (size_t)CCP;
    int P = (B > 0) ? (int)(ws_size / (bytes_per_partial * (size_t)B)) : 0;
    if (P > 256) P = 256;
    P = (P / 4) * 4;                      // K-tiling in steps of 4

    if (P >= 4) {
        float* partials = (float*)d_ws;
        hist_partial_kernel<<<dim3((unsigned)P, (unsigned)B), 256, 0, stream>>>(
            yp, y, partials, Np, P);
        wmma_reduce_kernel<<<(unsigned)B, 32, 0, stream>>>(partials, out, P);
    } else {
        zero_out_kernel<<<(out_size + 255) / 256, 256, 0, stream>>>(out, out_size);
        const int Pf = 64;
        hist_atomic_kernel<<<dim3((unsigned)Pf, (unsigned)B), 256, 0, stream>>>(
            yp, y, out, Np, Pf);
    }
}